// DNCMemory_1176821039590
// MI455X (gfx1250) — compile-verified
//
#include <hip/hip_runtime.h>
#include <hip/hip_bf16.h>
#include <math.h>

typedef __attribute__((ext_vector_type(2))) float v2f;
typedef __attribute__((ext_vector_type(4))) float v4f;
typedef __attribute__((ext_vector_type(8))) float v8f;

#define EPS_F 1e-5f
#define B_  64
#define N_  1024
#define W_  64
#define R_  4
// NW = 1

// ---------------------------------------------------------------------------
// Kernel 1: usage
// usage = (pu + (1-pu)*pww) * prod_r (1 - fg[r]*prw[r])
// ---------------------------------------------------------------------------
__global__ __launch_bounds__(256) void usage_kernel(
    const float* __restrict__ pww,      // [B,1,N]
    const float* __restrict__ fg,       // [B,R]
    const float* __restrict__ prw,      // [B,R,N]
    const float* __restrict__ pu,       // [B,N]
    float* __restrict__ usage_out)      // [B,N]
{
    int g = blockIdx.x * blockDim.x + threadIdx.x;   // 0 .. B*N-1
    int b = g >> 10;
    int n = g & (N_ - 1);
    float ww = pww[g];
    float u0 = pu[g];
    float u = u0 + (1.0f - u0) * ww;
    float phi = 1.0f;
#pragma unroll
    for (int r = 0; r < R_; ++r) {
        float f = fg[b * R_ + r];
        float w = prw[(b * R_ + r) * N_ + n];
        phi *= (1.0f - f * w);
    }
    usage_out[g] = u * phi;
}

// ---------------------------------------------------------------------------
// Kernel 2: allocation (per-batch bitonic sort of 1024 elements in LDS)
// ---------------------------------------------------------------------------
__global__ __launch_bounds__(1024) void alloc_kernel(
    const float* __restrict__ usage,    // [B,N]
    float* __restrict__ alloc_out)      // [B,N]
{
    __shared__ float sv[N_];   // nonusage, sorted descending
    __shared__ int   si[N_];   // original indices
    __shared__ float sp[N_];   // scan buffer

    int b = blockIdx.x;
    int t = threadIdx.x;

    float u = usage[b * N_ + t];
    u = EPS_F + (1.0f - EPS_F) * u;     // clamp
    sv[t] = 1.0f - u;                   // nonusage
    si[t] = t;
    __syncthreads();

    // bitonic sort: descending value, ties broken by ascending index
    for (int k = 2; k <= N_; k <<= 1) {
        for (int j = k >> 1; j > 0; j >>= 1) {
            int partner = t ^ j;
            if (partner > t) {
                float va = sv[t], vb = sv[partner];
                int   ia = si[t], ib = si[partner];
                bool aFirst = (va > vb) || (va == vb && ia < ib);
                bool desc = ((t & k) == 0);
                if (desc != aFirst) {
                    sv[t] = vb; sv[partner] = va;
                    si[t] = ib; si[partner] = ia;
                }
            }
            __syncthreads();
        }
    }

    // exclusive product scan of sorted_usage = 1 - sorted_nonusage
    sp[t] = (t == 0) ? 1.0f : (1.0f - sv[t - 1]);
    __syncthreads();
    for (int off = 1; off < N_; off <<= 1) {
        float other = (t >= off) ? sp[t - off] : 1.0f;
        float cur = sp[t];
        __syncthreads();
        sp[t] = cur * other;
        __syncthreads();
    }

    // sorted_alloc scattered back to original positions
    alloc_out[b * N_ + si[t]] = sv[t] * sp[t];
}

// ---------------------------------------------------------------------------
// Kernel 3: cosine content weights for 5 heads (4 read + 1 write) via
// V_WMMA_F32_16X16X4_F32, fused memory-row norm + softmax.
// Grid: B blocks, 256 threads (8 waves); each wave does 8 n-tiles of 16.
// ---------------------------------------------------------------------------
__device__ __forceinline__ float softplus_f(float x) {
    return fmaxf(x, 0.0f) + log1pf(expf(-fabsf(x)));
}

__global__ __launch_bounds__(256) void content_kernel(
    const float* __restrict__ mem,      // [B,N,W]
    const float* __restrict__ rk,       // [B,R,W]
    const float* __restrict__ rs,       // [B,R]
    const float* __restrict__ wk,       // [B,1,W]
    const float* __restrict__ wstr,     // [B,1]
    float* __restrict__ read_w,         // [B,R,N]  (d_out)
    float* __restrict__ write_c)        // [B,N]    (ws)
{
    __shared__ float s_sharp[5][N_];
    __shared__ float s_mn[N_];
    __shared__ float s_kn[5];
    __shared__ float s_sp[5];
    __shared__ float s_red[8];

    int b    = blockIdx.x;
    int tid  = threadIdx.x;
    int wave = tid >> 5;
    int lane = tid & 31;
    int hrow = lane & 15;
    int koff = (lane < 16) ? 0 : 2;     // K column pair for this half-wave

    // --- A fragment (keys), rows 0-3 = read keys, row 4 = write key, 5-15 = 0
    const float* kptr = nullptr;
    if (hrow < 4)       kptr = rk + ((size_t)b * R_ + hrow) * W_;
    else if (hrow == 4) kptr = wk + (size_t)b * W_;

    v2f afrag[16];
#pragma unroll
    for (int kk = 0; kk < 16; ++kk) {
        if (kptr) {
            afrag[kk].x = kptr[kk * 4 + koff];
            afrag[kk].y = kptr[kk * 4 + koff + 1];
        } else {
            afrag[kk].x = 0.0f; afrag[kk].y = 0.0f;
        }
    }

    // --- dot products + memory row norms, 16-row tiles
    for (int tile = wave; tile < N_ / 16; tile += 8) {
        int n0 = tile * 16;
        int nrow = n0 + hrow;
        const float* mrow = mem + ((size_t)b * N_ + nrow) * W_;

        v8f c = {};
        float nacc = 0.0f;
#pragma unroll
        for (int kk = 0; kk < 16; ++kk) {
            v2f bb;
            bb.x = mrow[kk * 4 + koff];
            bb.y = mrow[kk * 4 + koff + 1];
            nacc = fmaf(bb.x, bb.x, nacc);
            nacc = fmaf(bb.y, bb.y, nacc);
            c = __builtin_amdgcn_wmma_f32_16x16x4_f32(
                    false, afrag[kk], false, bb, (short)0, c, false, false);
        }
        // halves of the wave hold complementary K columns of the norm sum
        nacc += __shfl_xor(nacc, 16, 32);
        if (lane < 16) {
            s_mn[nrow] = sqrtf(nacc + EPS_F);
            s_sharp[0][nrow] = c[0];
            s_sharp[1][nrow] = c[1];
            s_sharp[2][nrow] = c[2];
            s_sharp[3][nrow] = c[3];
            s_sharp[4][nrow] = c[4];
        }
    }
    __syncthreads();

    // --- key norms + softplus(strength)
    if (tid < 5) {
        const float* kp = (tid < 4) ? rk + ((size_t)b * R_ + tid) * W_
                                    : wk + (size_t)b * W_;
        float s = 0.0f;
        for (int k = 0; k < W_; ++k) s = fmaf(kp[k], kp[k], s);
        s_kn[tid] = sqrtf(s + EPS_F);
        float st = (tid < 4) ? rs[b * R_ + tid] : wstr[b];
        s_sp[tid] = softplus_f(st);
    }
    __syncthreads();

    // --- per-head normalize + softmax over N
    for (int h = 0; h < 5; ++h) {
        float kn = s_kn[h];
        float sp = s_sp[h];
        float lmax = -INFINITY;
        for (int n = tid; n < N_; n += 256) {
            float v = s_sharp[h][n] / (kn * s_mn[n] + EPS_F) * sp;
            s_sharp[h][n] = v;
            lmax = fmaxf(lmax, v);
        }
#pragma unroll
        for (int o = 16; o > 0; o >>= 1) lmax = fmaxf(lmax, __shfl_xor(lmax, o, 32));
        if (lane == 0) s_red[wave] = lmax;
        __syncthreads();
        float gmax = s_red[0];
#pragma unroll
        for (int i = 1; i < 8; ++i) gmax = fmaxf(gmax, s_red[i]);
        __syncthreads();

        float lsum = 0.0f;
        for (int n = tid; n < N_; n += 256) {
            float e = expf(s_sharp[h][n] - gmax);
            s_sharp[h][n] = e;
            lsum += e;
        }
#pragma unroll
        for (int o = 16; o > 0; o >>= 1) lsum += __shfl_xor(lsum, o, 32);
        if (lane == 0) s_red[wave] = lsum;
        __syncthreads();
        float gsum = 0.0f;
#pragma unroll
        for (int i = 0; i < 8; ++i) gsum += s_red[i];
        float inv = 1.0f / gsum;
        for (int n = tid; n < N_; n += 256) {
            float w = s_sharp[h][n] * inv;
            if (h < 4) read_w[((size_t)b * R_ + h) * N_ + n] = w;
            else       write_c[(size_t)b * N_ + n] = w;
        }
        __syncthreads();
    }
}

// ---------------------------------------------------------------------------
// Kernel 4: write weights + precedence (per-batch block, sum reduction)
// ww = wg*(ag*alloc + (1-ag)*write_content)
// prec = (1 - sum(ww))*prev_prec + ww
// ---------------------------------------------------------------------------
__global__ __launch_bounds__(256) void combine_kernel(
    const float* __restrict__ allocA,   // [B,N] ws
    const float* __restrict__ wc,       // [B,N] ws
    const float* __restrict__ ag_,      // [B,1]
    const float* __restrict__ wg_,      // [B,1]
    const float* __restrict__ pprec,    // [B,1,N]
    float* __restrict__ ww_out,         // [B,1,N]
    float* __restrict__ prec_out)       // [B,1,N]
{
    __shared__ float s_red[8];
    int b = blockIdx.x;
    int tid = threadIdx.x;
    int wave = tid >> 5, lane = tid & 31;
    float ag = ag_[b], wg = wg_[b];

    float wwv[4];
    float lsum = 0.0f;
#pragma unroll
    for (int s = 0; s < 4; ++s) {
        int n = tid + s * 256;
        float w = wg * (ag * allocA[b * N_ + n] + (1.0f - ag) * wc[b * N_ + n]);
        wwv[s] = w;
        ww_out[b * N_ + n] = w;
        lsum += w;
    }
#pragma unroll
    for (int o = 16; o > 0; o >>= 1) lsum += __shfl_xor(lsum, o, 32);
    if (lane == 0) s_red[wave] = lsum;
    __syncthreads();
    float gsum = 0.0f;
#pragma unroll
    for (int i = 0; i < 8; ++i) gsum += s_red[i];
    float om = 1.0f - gsum;
#pragma unroll
    for (int s = 0; s < 4; ++s) {
        int n = tid + s * 256;
        prec_out[b * N_ + n] = om * pprec[b * N_ + n] + wwv[s];
    }
}

// ---------------------------------------------------------------------------
// Kernel 5: link update (bandwidth-dominant: 512 MB streamed)
// link[b,i,j] = (1 - ww[i] - ww[j])*prev_link[b,i,j] + ww[i]*prev_prec[j],
// diagonal zeroed. One block per (b,i) row; 256 threads x 128-bit vectors.
// prev_link / link are touched exactly once -> non-temporal (TH=NT) so the
// 512 MB stream doesn't evict the L2-resident ww/prec vectors (4 KB/batch,
// reused by 1024 row-blocks each).
// ---------------------------------------------------------------------------
__global__ __launch_bounds__(256) void link_kernel(
    const float* __restrict__ plink,    // [B,1,N,N]
    const float* __restrict__ ww,       // [B,1,N]
    const float* __restrict__ pprec,    // [B,1,N]
    float* __restrict__ link_out)       // [B,1,N,N]
{
    int row = blockIdx.x;               // b*N + i
    int b = row >> 10;
    int i = row & (N_ - 1);
    float wwi = ww[row];
    float omw = 1.0f - wwi;

    size_t rowbase = (size_t)row * N_;
    int j0 = threadIdx.x * 4;

    v4f pl4 = __builtin_nontemporal_load((const v4f*)(plink + rowbase + j0));
    v4f wj4 = *(const v4f*)(ww + (size_t)b * N_ + j0);
    v4f pr4 = *(const v4f*)(pprec + (size_t)b * N_ + j0);

    v4f o;
    o.x = (omw - wj4.x) * pl4.x + wwi * pr4.x;
    o.y = (omw - wj4.y) * pl4.y + wwi * pr4.y;
    o.z = (omw - wj4.z) * pl4.z + wwi * pr4.z;
    o.w = (omw - wj4.w) * pl4.w + wwi * pr4.w;
    if (i - j0 >= 0 && i - j0 < 4) {
        if (i - j0 == 0) o.x = 0.0f;
        else if (i - j0 == 1) o.y = 0.0f;
        else if (i - j0 == 2) o.z = 0.0f;
        else o.w = 0.0f;
    }
    __builtin_nontemporal_store(o, (v4f*)(link_out + rowbase + j0));
}

// ---------------------------------------------------------------------------
extern "C" void kernel_launch(void* const* d_in, const int* in_sizes, int n_in,
                              void* d_out, int out_size, void* d_ws, size_t ws_size,
                              hipStream_t stream) {
    const float* memory      = (const float*)d_in[0];
    const float* read_keys   = (const float*)d_in[1];
    const float* read_str    = (const float*)d_in[2];
    const float* write_keys  = (const float*)d_in[3];
    const float* write_str   = (const float*)d_in[4];
    const float* free_gate   = (const float*)d_in[5];
    const float* alloc_gate  = (const float*)d_in[6];
    const float* write_gate  = (const float*)d_in[7];
    const float* prev_rw     = (const float*)d_in[8];
    const float* prev_ww     = (const float*)d_in[9];
    const float* prev_usage  = (const float*)d_in[10];
    const float* prev_link   = (const float*)d_in[11];
    const float* prev_prec   = (const float*)d_in[12];

    float* out_read  = (float*)d_out;                       // [64,4,1024]
    float* out_ww    = out_read + (size_t)B_ * R_ * N_;     // [64,1,1024]
    float* out_usage = out_ww + (size_t)B_ * N_;            // [64,1024]
    float* out_link  = out_usage + (size_t)B_ * N_;         // [64,1,1024,1024]
    float* out_prec  = out_link + (size_t)B_ * N_ * N_;     // [64,1,1024]

    float* ws_alloc = (float*)d_ws;                         // [64,1024]
    float* ws_wc    = ws_alloc + (size_t)B_ * N_;           // [64,1024]

    usage_kernel<<<(B_ * N_) / 256, 256, 0, stream>>>(
        prev_ww, free_gate, prev_rw, prev_usage, out_usage);

    alloc_kernel<<<B_, N_, 0, stream>>>(out_usage, ws_alloc);

    content_kernel<<<B_, 256, 0, stream>>>(
        memory, read_keys, read_str, write_keys, write_str, out_read, ws_wc);

    combine_kernel<<<B_, 256, 0, stream>>>(
        ws_alloc, ws_wc, alloc_gate, write_gate, prev_prec, out_ww, out_prec);

    link_kernel<<<B_ * N_, 256, 0, stream>>>(
        prev_link, out_ww, prev_prec, out_link);
}